// Sampler_28767690949169
// MI455X (gfx1250) — compile-verified
//
#include <hip/hip_runtime.h>
#include <hip/hip_bf16.h>
#include <stdint.h>

// ---------------------------------------------------------------------------
// Top-k / top-p / min-p sampler for MI455X (gfx1250, wave32).
// One 1024-thread workgroup (32 waves) per batch row.
//   1) 4-pass radix select over float bit patterns (all probs >= 0, so
//      integer order == float order) to find the 1024th-largest value.
//      Histograms are per-wave + wave-aggregated (match-any) LDS atomics to
//      kill same-bucket ds_add serialization (probs are near-uniform, so
//      almost all elements share one exponent bucket in pass 0).
//   2) Gather pass: strictly-greater values appended via LDS atomics (rare:
//      ~K hits per row), remainder padded with the threshold value.
//   3) Bitonic sort (descending) of 1024 values in LDS.
//   4) Hillis-Steele exclusive prefix sum, masks, renormalize.
// Streaming passes are staged through LDS with double-buffered
// global_load_async_to_lds_b128 + s_wait_asynccnt (ASYNCcnt path).
// ---------------------------------------------------------------------------

#define NTHREADS 1024
#define WAVE     32            // gfx1250 is wave32-only
#define NWAVES   (NTHREADS / WAVE)
#define MAXK     1024
#define TILE_FLOATS (NTHREADS * 4)   // 16 KB per stage buffer

typedef int v4i __attribute__((ext_vector_type(4)));

#ifndef __has_builtin
#define __has_builtin(x) 0
#endif

#if __has_builtin(__builtin_amdgcn_global_load_async_to_lds_b128)
#define HAVE_ASYNC_LDS 1
#else
#define HAVE_ASYNC_LDS 0
#endif

#if __has_builtin(__builtin_amdgcn_s_wait_asynccnt)
#define WAIT_ASYNC(n) __builtin_amdgcn_s_wait_asynccnt(n)
#else
#define WAIT_ASYNC(n) asm volatile("s_wait_asynccnt %0" ::"i"(n) : "memory")
#endif

// Wave-aggregated histogram add (match-any): lanes holding the same digit
// elect a leader which adds popcount once. Iterations == #distinct digits
// among the wave's pending lanes (typically 1-3 for near-uniform probs).
__device__ __forceinline__ void hist_add_agg(unsigned* whist, unsigned digit,
                                             bool pred) {
  const int lane = (int)(threadIdx.x & (WAVE - 1));
  bool pending = pred;
  while (true) {
    unsigned long long ball = __ballot((int)pending);
    if (ball == 0ull) break;
    const int leader = __ffsll(ball) - 1;
    const unsigned ld = (unsigned)__shfl((int)digit, leader, WAVE);
    const bool same = pending && (digit == ld);
    unsigned long long sm = __ballot((int)same);
    if (lane == leader) atomicAdd(&whist[ld], (unsigned)__popcll(sm));
    pending = pending && !same;
  }
}

// Stream one row of V floats through fn(float). Uniform control flow across
// the workgroup (barriers inside the async path).
template <typename F>
__device__ __forceinline__ void stream_row(const float* __restrict__ row, int V,
                                           float (*stage)[TILE_FLOATS], F fn) {
  const int tid = threadIdx.x;
#if HAVE_ASYNC_LDS
  const int ntiles = V / TILE_FLOATS;
  if (ntiles > 0) {
    __builtin_amdgcn_global_load_async_to_lds_b128(
        (v4i*)(row + tid * 4),
        (__attribute__((address_space(3))) v4i*)&stage[0][tid * 4], 0, 0);
  }
  for (int t = 0; t < ntiles; ++t) {
    if (t + 1 < ntiles) {
      const float* g = row + (size_t)(t + 1) * TILE_FLOATS + tid * 4;
      __builtin_amdgcn_global_load_async_to_lds_b128(
          (v4i*)g,
          (__attribute__((address_space(3))) v4i*)&stage[(t + 1) & 1][tid * 4],
          0, 0);
      WAIT_ASYNC(1);   // two in flight per lane; oldest (tile t) has landed
    } else {
      WAIT_ASYNC(0);
    }
    __syncthreads();   // tile t visible from every wave
    float4 v = ((const float4*)stage[t & 1])[tid];
    fn(v.x); fn(v.y); fn(v.z); fn(v.w);
    __syncthreads();   // done reading before this buffer is overwritten
  }
  for (int i = ntiles * TILE_FLOATS + tid; i < V; i += NTHREADS) fn(row[i]);
#else
  (void)stage;
  const int nvec = V >> 2;
  const float4* rv = (const float4*)row;
  for (int i = tid; i < nvec; i += NTHREADS) {
    __builtin_prefetch(rv + i + 4 * NTHREADS, 0, 1);
    float4 v = rv[i];
    fn(v.x); fn(v.y); fn(v.z); fn(v.w);
  }
  for (int i = (nvec << 2) + tid; i < V; i += NTHREADS) fn(row[i]);
#endif
}

__global__ __launch_bounds__(NTHREADS) void
sampler_topk_topp_minp(const float* __restrict__ probs,
                       const float* __restrict__ top_p,
                       const int* __restrict__ top_k,
                       const float* __restrict__ min_p,
                       const unsigned char* __restrict__ apply_min_p,
                       float* __restrict__ out, int V, int K) {
  const int row = blockIdx.x;
  const int tid = threadIdx.x;
  const float* rowp = probs + (size_t)row * (size_t)V;

#if HAVE_ASYNC_LDS
  __shared__ float s_stage[2][TILE_FLOATS];
  float(*stagep)[TILE_FLOATS] = s_stage;
#else
  float(*stagep)[TILE_FLOATS] = nullptr;
#endif
  __shared__ unsigned s_whist[NWAVES][256];   // per-wave histograms (32 KB)
  __shared__ float    s_vals[MAXK];
  __shared__ float    s_scan[NTHREADS];
  __shared__ unsigned s_hist[256];            // merged histogram
  __shared__ unsigned s_prefix, s_remaining, s_cnt;
  __shared__ float    s_minp_thr;

  unsigned* myhist = s_whist[tid >> 5];       // this wave's histogram row

  // ---- 1) radix select: bit pattern of the K-th largest value -------------
  if (tid == 0) { s_prefix = 0u; s_remaining = (unsigned)K; }
  __syncthreads();

  for (int pass = 0; pass < 4; ++pass) {
    const int shift = 24 - 8 * pass;
    // clear all per-wave histograms (32*256 entries / 1024 threads = 8 each)
    for (int i = tid; i < NWAVES * 256; i += NTHREADS)
      ((unsigned*)s_whist)[i] = 0u;
    __syncthreads();
    const unsigned prefix  = s_prefix;
    const unsigned hi_mask = (pass == 0) ? 0u : (0xFFFFFFFFu << (shift + 8));
    stream_row(rowp, V, stagep, [&](float f) {
      unsigned u = __float_as_uint(f);
      hist_add_agg(myhist, (u >> shift) & 0xFFu, (u & hi_mask) == prefix);
    });
    __syncthreads();
    if (tid < 256) {                          // merge wave histograms
      unsigned c = 0;
      for (int w = 0; w < NWAVES; ++w) c += s_whist[w][tid];
      s_hist[tid] = c;
    }
    __syncthreads();
    if (tid == 0) {
      unsigned rem = s_remaining, acc = 0;
      int d = 255;
      for (; d > 0; --d) {
        unsigned c = s_hist[d];
        if (acc + c >= rem) break;
        acc += c;
      }
      s_prefix    = prefix | ((unsigned)d << shift);
      s_remaining = rem - acc;   // # of threshold-equal values still needed
    }
    __syncthreads();
  }

  // ---- 2) gather: strictly-greater values, pad with threshold -------------
  if (tid == 0) s_cnt = 0u;
  __syncthreads();
  const unsigned thr = s_prefix;
  stream_row(rowp, V, stagep, [&](float f) {
    if (__float_as_uint(f) > thr) {          // ~K hits per row: low contention
      unsigned p = atomicAdd(&s_cnt, 1u);
      if (p < (unsigned)K) s_vals[p] = f;
    }
  });
  __syncthreads();
  {
    const unsigned cgt = s_cnt;              // == K - s_remaining
    const float thrF = __uint_as_float(thr);
    for (int i = tid; i < K; i += NTHREADS)
      if ((unsigned)i >= cgt) s_vals[i] = thrF;
  }
  __syncthreads();

  // ---- 3) bitonic sort, descending (K is a power of two <= 1024) ----------
  for (int k2 = 2; k2 <= K; k2 <<= 1) {
    for (int j = k2 >> 1; j > 0; j >>= 1) {
      const int i = tid, ixj = tid ^ j;
      if (i < K && ixj > i) {
        float a = s_vals[i], b = s_vals[ixj];
        const bool desc = ((i & k2) == 0);
        if (desc ? (a < b) : (a > b)) { s_vals[i] = b; s_vals[ixj] = a; }
      }
      __syncthreads();
    }
  }

  // ---- 4) exclusive prefix sum (Hillis-Steele) -----------------------------
  const float x = (tid < K) ? s_vals[tid] : 0.0f;
  s_scan[tid] = x;
  __syncthreads();
  for (int off = 1; off < K; off <<= 1) {
    float t = (tid >= off) ? s_scan[tid - off] : 0.0f;
    __syncthreads();
    s_scan[tid] += t;
    __syncthreads();
  }
  const float excl = s_scan[tid] - x;   // cumsum - probs_sort
  __syncthreads();

  // ---- masks + normalize ---------------------------------------------------
  const float tp = top_p[row];
  const int   tk = top_k[row];
  const bool  ap = apply_min_p[row] != 0;
  if (tid == 0) {
    const bool first_passes = (tk > 0) && !(0.0f > tp);
    s_minp_thr = (ap && first_passes) ? s_vals[0] * min_p[row] : 0.0f;
  }
  __syncthreads();

  const bool keep = (tid < K) && (tid < tk) && !(excl > tp) &&
                    !(ap && (x < s_minp_thr));
  const float filt = keep ? x : 0.0f;

  s_scan[tid] = filt;
  __syncthreads();
  for (int s = NTHREADS >> 1; s > 0; s >>= 1) {
    if (tid < s) s_scan[tid] += s_scan[tid + s];
    __syncthreads();
  }
  const float sum = s_scan[0];
  if (tid < K) out[(size_t)row * (size_t)K + tid] = filt / sum;
}

extern "C" void kernel_launch(void* const* d_in, const int* in_sizes, int n_in,
                              void* d_out, int out_size, void* d_ws, size_t ws_size,
                              hipStream_t stream) {
  (void)n_in; (void)d_ws; (void)ws_size;
  const float*         probs = (const float*)d_in[0];
  const float*         top_p = (const float*)d_in[1];
  const int*           top_k = (const int*)d_in[2];
  const float*         min_p = (const float*)d_in[3];
  const unsigned char* apply = (const unsigned char*)d_in[4];
  // d_in[5] is max_k as a device scalar; derive K on host instead.
  const int B = in_sizes[1];           // top_p has one entry per row
  const int V = in_sizes[0] / B;       // 128000
  const int K = out_size / B;          // 1024 (power of two, <= MAXK)
  sampler_topk_topp_minp<<<B, NTHREADS, 0, stream>>>(probs, top_p, top_k, min_p,
                                                     apply, (float*)d_out, V, K);
}